// HOSVDAggregator_48842368090367
// MI455X (gfx1250) — compile-verified
//
#include <hip/hip_runtime.h>

// HOSVD aggregator: proj (WMMA K=256) -> fused Tucker core+b-reduction (WMMA)
// -> output GEMM (WMMA). All matrix math on v_wmma_f32_16x16x4_f32 (exact fp32).
// 512-thread workgroups (16 waves) so one ~292KB-LDS workgroup still yields
// 4 waves/SIMD on the WGP for latency hiding.

#define NN     200000
#define TILES  (NN / 16)   // 12500, exact
#define WGSZ   512
#define NWAVES 16

typedef float v2f __attribute__((ext_vector_type(2)));
typedef float v8f __attribute__((ext_vector_type(8)));

// LDS layout (floats)
#define GP_OFF   0        // G permuted: [(a>>1)][b][c][a&1]      32768 f
#define UWT_OFF  32768    // UwT: [d][(i>>1)][r][i&1]             16384 f
#define UOT_OFF  49152    // UoT: [(c>>1)][o][c&1]                 8192 f
#define UB_OFF   57344    // U_b raw                                 64 f
#define SCR_OFF  57408    // per-wave scratch: 2 x (16 rows x stride 34)
#define PSTR     34       // padded row stride (even => 8B-aligned b64 loads)
#define SCR_PW   (2 * 16 * PSTR)               // 1088 f per wave
#define LDS_F    (SCR_OFF + NWAVES * SCR_PW)   // 74816 f = 292.25 KB

__global__ __launch_bounds__(WGSZ, 1) void hosvd_kernel(
    const float* __restrict__ ns,   // [N,2,256]
    const float* __restrict__ G,    // [32,32,32]
    const float* __restrict__ Uw,   // [2,32,256]
    const float* __restrict__ Ub,   // [2,32]
    const float* __restrict__ Uo,   // [256,32]
    float* __restrict__ out)        // [N,256]
{
  __shared__ __align__(16) float lds[LDS_F];
  const int tid = threadIdx.x;

  // ---- cooperative staging of all B-operand matrices into pair layouts ----
  for (int idx = tid; idx < 32768; idx += WGSZ) {          // G[a][b][c]
    int a = idx >> 10, b = (idx >> 5) & 31, c = idx & 31;
    lds[GP_OFF + (a >> 1) * 2048 + b * 64 + c * 2 + (a & 1)] = G[idx];
  }
  for (int idx = tid; idx < 16384; idx += WGSZ) {          // Uw[d][r][i]
    int d = idx >> 13, r = (idx >> 8) & 31, i = idx & 255;
    lds[UWT_OFF + d * 8192 + (i >> 1) * 64 + r * 2 + (i & 1)] = Uw[idx];
  }
  for (int idx = tid; idx < 8192; idx += WGSZ) {           // Uo[o][c]
    int o = idx >> 5, c = idx & 31;
    lds[UOT_OFF + (c >> 1) * 512 + o * 2 + (c & 1)] = Uo[idx];
  }
  if (tid < 64) lds[UB_OFF + tid] = Ub[tid];
  __syncthreads();

  const int lane = tid & 31;
  const int hh   = lane >> 4;   // half-wave select (K rows 0,1 vs 2,3)
  const int lcol = lane & 15;   // column within 16-wide tile
  const int wave = tid >> 5;
  float* scrA = &lds[SCR_OFF + wave * SCR_PW];
  float* scrB = scrA + 16 * PSTR;

  const int wglobal = blockIdx.x * NWAVES + wave;
  const int nwaves  = gridDim.x * NWAVES;

  for (int t = wglobal; t < TILES; t += nwaves) {
    const int base = t * 16;

    // ---------- projections p0,p1 : [16x256] @ UwT -> [16x32] (+bias) ------
    v8f p[2][2];
#pragma unroll
    for (int d = 0; d < 2; ++d) {
      const float b0v = lds[UB_OFF + d * 32 + lcol];
      const float b1v = lds[UB_OFF + d * 32 + 16 + lcol];
      v8f c0, c1;
#pragma unroll
      for (int i = 0; i < 8; ++i) { c0[i] = b0v; c1[i] = b1v; }
      const float* abase = ns + ((size_t)(base + lcol) * 2 + d) * 256 + hh * 2;
      const float* wbase = &lds[UWT_OFF + d * 8192 + hh * 64 + lcol * 2];
#pragma unroll 4
      for (int k0 = 0; k0 < 256; k0 += 4) {
        v2f a  = *(const v2f*)(abase + k0);
        v2f b0 = *(const v2f*)(wbase + (k0 >> 1) * 64);
        v2f b1 = *(const v2f*)(wbase + (k0 >> 1) * 64 + 32);
        c0 = __builtin_amdgcn_wmma_f32_16x16x4_f32(false, a, false, b0, (short)0, c0, false, false);
        c1 = __builtin_amdgcn_wmma_f32_16x16x4_f32(false, a, false, b1, (short)0, c1, false, false);
      }
      p[d][0] = c0; p[d][1] = c1;
    }

    // ---- transpose p0 -> A layout, stash p1 rows, via wave-private LDS ----
#pragma unroll
    for (int v = 0; v < 8; ++v) {
      scrA[(v + hh * 8) * PSTR + lcol]      = p[0][0][v];
      scrA[(v + hh * 8) * PSTR + 16 + lcol] = p[0][1][v];
      scrB[(v + hh * 8) * PSTR + lcol]      = p[1][0][v];
      scrB[(v + hh * 8) * PSTR + 16 + lcol] = p[1][1][v];
    }
    asm volatile("s_wait_dscnt 0" ::: "memory");
    v2f pA[8];
#pragma unroll
    for (int k = 0; k < 8; ++k)
      pA[k] = *(const v2f*)(scrA + lcol * PSTR + k * 4 + hh * 2);

    // -------- fused Tucker core + b-reduction:
    // r[n,c] = sum_b sum_a (p1[n,b]*p0[n,a]) * G[a,b,c]  (b-scale folded in A)
    v8f r0, r1;
#pragma unroll
    for (int i = 0; i < 8; ++i) { r0[i] = 0.f; r1[i] = 0.f; }
    const float* gpb   = &lds[GP_OFF + hh * 2048 + lcol * 2];
    const float* p1row = scrB + lcol * PSTR;
    for (int b = 0; b < 32; ++b) {
      const float s = p1row[b];
      const float* gpbb = gpb + b * 64;
#pragma unroll
      for (int k = 0; k < 8; ++k) {
        v2f ap; ap[0] = pA[k][0] * s; ap[1] = pA[k][1] * s;
        v2f g0 = *(const v2f*)(gpbb + k * 4096);
        v2f g1 = *(const v2f*)(gpbb + k * 4096 + 32);
        r0 = __builtin_amdgcn_wmma_f32_16x16x4_f32(false, ap, false, g0, (short)0, r0, false, false);
        r1 = __builtin_amdgcn_wmma_f32_16x16x4_f32(false, ap, false, g1, (short)0, r1, false, false);
      }
    }

    // ---------------- transpose r -> A layout ----------------
#pragma unroll
    for (int v = 0; v < 8; ++v) {
      scrA[(v + hh * 8) * PSTR + lcol]      = r0[v];
      scrA[(v + hh * 8) * PSTR + 16 + lcol] = r1[v];
    }
    asm volatile("s_wait_dscnt 0" ::: "memory");
    v2f rA[8];
#pragma unroll
    for (int k = 0; k < 8; ++k)
      rA[k] = *(const v2f*)(scrA + lcol * PSTR + k * 4 + hh * 2);

    // ---------- output GEMM: [16x32] @ UoT -> [16x256] ----------
    float* obase = out + (size_t)base * 256;
    const float* uob = &lds[UOT_OFF + hh * 512 + lcol * 2];
    for (int tt = 0; tt < 16; ++tt) {
      v8f c;
#pragma unroll
      for (int i = 0; i < 8; ++i) c[i] = 0.f;
#pragma unroll
      for (int k = 0; k < 8; ++k) {
        v2f bo = *(const v2f*)(uob + k * 1024 + tt * 32);
        c = __builtin_amdgcn_wmma_f32_16x16x4_f32(false, rA[k], false, bo, (short)0, c, false, false);
      }
#pragma unroll
      for (int v = 0; v < 8; ++v)
        obase[(size_t)(v + hh * 8) * 256 + tt * 16 + lcol] = c[v];
    }
  }
}

extern "C" void kernel_launch(void* const* d_in, const int* in_sizes, int n_in,
                              void* d_out, int out_size, void* d_ws, size_t ws_size,
                              hipStream_t stream) {
  (void)in_sizes; (void)n_in; (void)d_ws; (void)ws_size; (void)out_size;
  const float* ns = (const float*)d_in[0];   // [N,2,256]
  const float* G  = (const float*)d_in[1];   // [32,32,32]
  const float* Uw = (const float*)d_in[2];   // [2,32,256]
  const float* Ub = (const float*)d_in[3];   // [2,32]
  const float* Uo = (const float*)d_in[4];   // [256,32]
  float* out = (float*)d_out;                // [N,256]
  hosvd_kernel<<<256, WGSZ, 0, stream>>>(ns, G, Uw, Ub, Uo, out);
}